// PPCA_16338055594484
// MI455X (gfx1250) — compile-verified
//
#include <hip/hip_runtime.h>
#include <hip/hip_bf16.h>
#include <math.h>

typedef __attribute__((ext_vector_type(2))) float v2f;
typedef __attribute__((ext_vector_type(8))) float v8f;

#define DDIM 64
#define KDIM 16
#define ROWS_PER_BLOCK 256   // samples per block in the Gram pass
#define INV_BETA 0.1f        // 1/BETA
#define LOG2PI 1.8378770664093453f

// ---------------------------------------------------------------------------
// Kernel 0: zero the Gram accumulator (in d_ws) and the scalar output.
// ---------------------------------------------------------------------------
__global__ void ppca_zero_kernel(float* __restrict__ G, float* __restrict__ out) {
    int i = blockIdx.x * blockDim.x + threadIdx.x;
    if (i < DDIM * DDIM) G[i] = 0.0f;
    if (i == 0) out[0] = 0.0f;
}

// ---------------------------------------------------------------------------
// Kernel 1: G = x^T x  via V_WMMA_F32_16X16X4_F32.
// x is (n, 64) row-major f32. Each block handles ROWS_PER_BLOCK samples.
// 8 waves per block; wave w computes tiles (I = w>>1, J in {2*(w&1), 2*(w&1)+1}).
// f32 16x4 A layout: lane L<16 -> v0=K0, v1=K1 at M=L; lane L>=16 -> K2,K3 at
// M=L-16.  B (4x16) uses the same striping with N in place of M, so the same
// per-lane load pattern (two rows of one column) serves both operands.
// ---------------------------------------------------------------------------
__global__ void ppca_gram_kernel(const float* __restrict__ x,
                                 float* __restrict__ G) {
    const int lane = threadIdx.x & 31;
    const int wave = threadIdx.x >> 5;          // 0..7
    const int I    = wave >> 1;                 // row block 0..3
    const int J0   = (wave & 1) * 2;            // col blocks J0, J0+1
    const int half = lane >> 4;                 // 0: K=0,1   1: K=2,3
    const int lid  = lane & 15;                 // M / N within tile

    size_t n0 = (size_t)blockIdx.x * ROWS_PER_BLOCK;
    const float* xa  = x + (n0 + 2 * half) * DDIM + I  * 16      + lid;
    const float* xb0 = x + (n0 + 2 * half) * DDIM + J0 * 16      + lid;
    const float* xb1 = x + (n0 + 2 * half) * DDIM + (J0 + 1) * 16 + lid;

    v8f acc0 = {};
    v8f acc1 = {};

#pragma unroll 4
    for (int it = 0; it < ROWS_PER_BLOCK; it += 4) {
        v2f a, b0, b1;
        a[0]  = xa[0];   a[1]  = xa[DDIM];      // rows n0+2h, n0+2h+1
        b0[0] = xb0[0];  b0[1] = xb0[DDIM];
        b1[0] = xb1[0];  b1[1] = xb1[DDIM];
        // D = A x B + C   (16x16x4, f32)
        acc0 = __builtin_amdgcn_wmma_f32_16x16x4_f32(
            false, a, false, b0, (short)0, acc0, false, false);
        acc1 = __builtin_amdgcn_wmma_f32_16x16x4_f32(
            false, a, false, b1, (short)0, acc1, false, false);
        xa  += 4 * DDIM;
        xb0 += 4 * DDIM;
        xb1 += 4 * DDIM;
    }

    // C/D layout: VGPR v, lanes 0-15 -> M=v, lanes 16-31 -> M=v+8, N=lane%16.
    const int rowbase = I * 16 + half * 8;
    const int col0 = J0 * 16 + lid;
    const int col1 = (J0 + 1) * 16 + lid;
#pragma unroll
    for (int v = 0; v < 8; ++v) {
        atomicAdd(&G[(rowbase + v) * DDIM + col0], acc0[v]);
        atomicAdd(&G[(rowbase + v) * DDIM + col1], acc1[v]);
    }
}

// ---------------------------------------------------------------------------
// Kernel 2: per-(a,p) epilogue.  One 64-thread block per task (128 tasks).
//   S  = Wp Wp^T + I/beta           (64x64, LDS)
//   Gp = G[perm][:,perm]            (64x64, LDS)
//   Gauss-Jordan invert S_rr in place (SPD, no pivoting) -> Inv
//   A  = S_mr @ Inv                 (m x r)
//   v_j = S[j,j] - sum_t A[j,t] S[j,m+t]
//   q_j = Gp[j,j] - 2 sum_t A[j,t] Gp[j,m+t] + sum_{t,u} A[j,t]A[j,u]Gp[m+t,m+u]
//   contrib = sum_j [ -0.5(log v_j + log 2pi) - q_j/(2 v_j n) ] / (P*m)
//   out -= contrib   (atomic)
// ---------------------------------------------------------------------------
__global__ void ppca_finalize_kernel(const float* __restrict__ W,
                                     const int* __restrict__ perms,
                                     const float* __restrict__ G,
                                     float* __restrict__ out,
                                     float fn) {
    __shared__ float S[DDIM * DDIM];    // 16 KB
    __shared__ float Gp[DDIM * DDIM];   // 16 KB
    __shared__ float Inv[DDIM * DDIM];  // 16 KB (rows 0..62 used; row 63 = perm)
    __shared__ float Aa[DDIM * DDIM];   // 16 KB
    int* pm = (int*)&Inv[63 * DDIM];    // alias: 64 ints in Inv's unused row

    const int task = blockIdx.x;        // 0..127
    const int a = task >> 1;
    const int p = task & 1;
    const int m = a + 1;
    const int r = DDIM - m;
    const int tid = threadIdx.x;        // 0..63

    pm[tid] = perms[(a * 2 + p) * DDIM + tid];
    __syncthreads();

    // Build row tid of S and Gp.
    {
        const int i = tid;
        const int pi = pm[i];
        float wi[KDIM];
#pragma unroll
        for (int k = 0; k < KDIM; ++k) wi[k] = W[pi * KDIM + k];
        for (int j = 0; j < DDIM; ++j) {
            const int pj = pm[j];
            float s = 0.0f;
#pragma unroll
            for (int k = 0; k < KDIM; ++k) s += wi[k] * W[pj * KDIM + k];
            if (i == j) s += INV_BETA;
            S[i * DDIM + j] = s;
            Gp[i * DDIM + j] = G[pi * DDIM + pj];
        }
    }
    __syncthreads();

    // Identity for Inv (r x r).
    if (tid < r) {
        for (int j = 0; j < r; ++j)
            Inv[tid * DDIM + j] = (tid == j) ? 1.0f : 0.0f;
    }
    __syncthreads();

    // Gauss-Jordan on the rr block of S, in place (SPD => no pivoting).
    // Srr(i,j) = S[(m+i)*64 + (m+j)].
    for (int k = 0; k < r; ++k) {
        const float pinv = 1.0f / S[(m + k) * DDIM + (m + k)];
        __syncthreads();
        if (tid < r) {
            S[(m + k) * DDIM + (m + tid)] *= pinv;
            Inv[k * DDIM + tid] *= pinv;
        }
        __syncthreads();
        if (tid < r && tid != k) {
            const float f = S[(m + tid) * DDIM + (m + k)];
            for (int j = 0; j < r; ++j) {
                S[(m + tid) * DDIM + (m + j)] -= f * S[(m + k) * DDIM + (m + j)];
                Inv[tid * DDIM + j]           -= f * Inv[k * DDIM + j];
            }
        }
        __syncthreads();
    }

    // A = S_mr @ Inv   (rows < m of S are untouched by the elimination).
    if (tid < m) {
        for (int j = 0; j < r; ++j) {
            float s = 0.0f;
            for (int t = 0; t < r; ++t)
                s += S[tid * DDIM + (m + t)] * Inv[t * DDIM + j];
            Aa[tid * DDIM + j] = s;
        }
    }
    __syncthreads();

    // Per-column v_j, q_j and the task contribution.
    float contrib = 0.0f;
    if (tid < m) {
        const int i = tid;
        float v = S[i * DDIM + i];
        for (int t = 0; t < r; ++t)
            v -= Aa[i * DDIM + t] * S[i * DDIM + (m + t)];
        float q = Gp[i * DDIM + i];
        for (int t = 0; t < r; ++t)
            q -= 2.0f * Aa[i * DDIM + t] * Gp[i * DDIM + (m + t)];
        for (int t = 0; t < r; ++t) {
            float s = 0.0f;
            for (int u = 0; u < r; ++u)
                s += Aa[i * DDIM + u] * Gp[(m + t) * DDIM + (m + u)];
            q += Aa[i * DDIM + t] * s;
        }
        contrib = -0.5f * (logf(v) + LOG2PI) - q / (2.0f * v * fn);
    }
    __syncthreads();            // S no longer needed; reuse row 0 for reduction
    S[tid] = contrib;
    __syncthreads();
    for (int s = 32; s > 0; s >>= 1) {
        if (tid < s) S[tid] += S[tid + s];
        __syncthreads();
    }
    if (tid == 0) {
        // total += contrib / (P*m); result is -total.
        atomicAdd(out, -S[0] / (2.0f * (float)m));
    }
}

// ---------------------------------------------------------------------------
extern "C" void kernel_launch(void* const* d_in, const int* in_sizes, int n_in,
                              void* d_out, int out_size, void* d_ws, size_t ws_size,
                              hipStream_t stream) {
    const float* x     = (const float*)d_in[0];  // (n, 64) f32
    const float* W     = (const float*)d_in[1];  // (64, 16) f32
    const int*   perms = (const int*)d_in[2];    // (64, 2, 64) i32
    float* out = (float*)d_out;                  // scalar f32
    float* G   = (float*)d_ws;                   // 64*64 f32 Gram accumulator

    const int n = in_sizes[0] / DDIM;            // 65536

    ppca_zero_kernel<<<16, 256, 0, stream>>>(G, out);

    const int gblocks = n / ROWS_PER_BLOCK;      // 256 for n=65536
    ppca_gram_kernel<<<gblocks, 256, 0, stream>>>(x, G);

    ppca_finalize_kernel<<<DDIM * 2, DDIM, 0, stream>>>(W, perms, G, out, (float)n);
}